// GraphConvolutionCollector_5583457485051
// MI455X (gfx1250) — compile-verified
//
#include <hip/hip_runtime.h>

// ---------------------------------------------------------------------------
// out[g,p] = bilinear_resize(x1 @ x2^T, n->64)  ==  (R x1) @ (R x2)^T
// where R is the 64xNMAX bilinear row-interp matrix (2 nnz/row, same for
// rows & cols). Per pair: gather/interp A'=R x1, B'=R x2 (64x128 each) into
// LDS, then a 64x64x128 fp32 GEMM via V_WMMA_F32_16X16X4_F32 (exact fp32,
// matching the fp32 reference).
// ---------------------------------------------------------------------------

typedef __attribute__((ext_vector_type(2))) float v2f;
typedef __attribute__((ext_vector_type(8))) float v8f;

#define G_   3
#define P_   512
#define NMAX_ 256
#define D_   128
#define FIX_ 64
#define LDSS 132   // 128 + 4 pad: conflict-free ds_load_b64 across a wave32,
                   // row stride 528 B keeps 16 B alignment for float4 stores

__global__ __launch_bounds__(256) void gcc_fused_wmma_kernel(
    const float* __restrict__ x,      // [G,P,2,NMAX,D] fp32
    const int*   __restrict__ sizes,  // [P,2] int32
    float*       __restrict__ out)    // [G,P,1,64,64] fp32
{
    __shared__ __align__(16) float Ash[FIX_ * LDSS];
    __shared__ __align__(16) float Bsh[FIX_ * LDSS];
    __shared__ float wsh[FIX_];
    __shared__ int   i0sh[FIX_];
    __shared__ int   i1sh[FIX_];

    const int bid = blockIdx.x;           // pair index: g*P + p
    const int p   = bid % P_;
    const int tid = threadIdx.x;

    // ---- phase 1: interpolation coords (align_corners=False, clamp>=0) ----
    const int n1 = sizes[p * 2 + 0];
    const int n2 = sizes[p * 2 + 1];
    const int n  = (n1 > n2) ? n1 : n2;
    if (tid < FIX_) {
        const float scale = (float)n * (1.0f / (float)FIX_);
        float src = ((float)tid + 0.5f) * scale - 0.5f;
        src = fmaxf(src, 0.0f);
        const int i0 = (int)floorf(src);
        const int i1 = min(i0 + 1, n - 1);
        wsh[tid]  = src - (float)i0;
        i0sh[tid] = i0;
        i1sh[tid] = i1;
    }
    __syncthreads();

    // ---- phase 2: gather + row-interp A' = R x1, B' = R x2 into LDS ------
    const float* x1 = x + (size_t)bid * (2 * NMAX_ * D_);
    const float* x2 = x1 + NMAX_ * D_;
    // 64 rows * 32 float4 = 2048 float4 per matrix; 256 threads -> 8 iters.
    // Threads of one iteration cover whole rows -> fully coalesced B128 loads.
    for (int it = 0; it < 8; ++it) {
        const int idx = tid + it * 256;
        const int r   = idx >> 5;          // output row 0..63
        const int c4  = (idx & 31) << 2;   // column 0..124 (float4 granules)
        const float w  = wsh[r];
        const float wi = 1.0f - w;
        const int r0 = i0sh[r], r1 = i1sh[r];

        const float4 a0 = *(const float4*)(x1 + (size_t)r0 * D_ + c4);
        const float4 a1 = *(const float4*)(x1 + (size_t)r1 * D_ + c4);
        const float4 b0 = *(const float4*)(x2 + (size_t)r0 * D_ + c4);
        const float4 b1 = *(const float4*)(x2 + (size_t)r1 * D_ + c4);

        float4 av, bv;
        av.x = wi * a0.x + w * a1.x;  av.y = wi * a0.y + w * a1.y;
        av.z = wi * a0.z + w * a1.z;  av.w = wi * a0.w + w * a1.w;
        bv.x = wi * b0.x + w * b1.x;  bv.y = wi * b0.y + w * b1.y;
        bv.z = wi * b0.z + w * b1.z;  bv.w = wi * b0.w + w * b1.w;

        *(float4*)(&Ash[r * LDSS + c4]) = av;   // row base 528B = 33*16 ✓
        *(float4*)(&Bsh[r * LDSS + c4]) = bv;
    }
    __syncthreads();

    // ---- phase 3: out = A' @ B'^T, 4x4 grid of 16x16 tiles, K=128 --------
    // 8 waves; wave handles (tileM, tileN0) and (tileM, tileN0+1), sharing
    // the A fragment between the two WMMAs each K-step.
    const int wave   = tid >> 5;
    const int lane   = tid & 31;
    const int half   = lane >> 4;      // 0: K pair {0,1}, 1: K pair {2,3}
    const int m      = lane & 15;      // M for A frag / N for B frag
    const int tileM  = wave >> 1;
    const int tileN0 = (wave & 1) * 2;

    const float* arow  = &Ash[(tileM * 16 + m)        * LDSS + 2 * half];
    const float* brow0 = &Bsh[(tileN0 * 16 + m)       * LDSS + 2 * half];
    const float* brow1 = &Bsh[((tileN0 + 1) * 16 + m) * LDSS + 2 * half];

    v8f acc0 = {0.f, 0.f, 0.f, 0.f, 0.f, 0.f, 0.f, 0.f};
    v8f acc1 = {0.f, 0.f, 0.f, 0.f, 0.f, 0.f, 0.f, 0.f};

#pragma unroll 4
    for (int k0 = 0; k0 < D_; k0 += 4) {
        v2f a, b0v, b1v;                       // ds_load_b64 each
        a.x   = arow[k0];   a.y   = arow[k0 + 1];
        b0v.x = brow0[k0];  b0v.y = brow0[k0 + 1];
        b1v.x = brow1[k0];  b1v.y = brow1[k0 + 1];
        acc0 = __builtin_amdgcn_wmma_f32_16x16x4_f32(
                   false, a, false, b0v, (short)0, acc0, false, false);
        acc1 = __builtin_amdgcn_wmma_f32_16x16x4_f32(
                   false, a, false, b1v, (short)0, acc1, false, false);
    }

    // C/D layout: VGPR j -> M = j (lanes 0-15) / j+8 (lanes 16-31), N = lane&15
    float* outp = out + (size_t)bid * (FIX_ * FIX_);
#pragma unroll
    for (int j = 0; j < 8; ++j) {
        const int row = tileM * 16 + j + 8 * half;
        outp[row * FIX_ + tileN0 * 16 + m]       = acc0[j];
        outp[row * FIX_ + (tileN0 + 1) * 16 + m] = acc1[j];
    }
}

extern "C" void kernel_launch(void* const* d_in, const int* in_sizes, int n_in,
                              void* d_out, int out_size, void* d_ws, size_t ws_size,
                              hipStream_t stream) {
    const float* x     = (const float*)d_in[0];   // [G,P,2,NMAX,D] fp32
    const int*   sizes = (const int*)d_in[1];     // [P,2] int32
    float*       out   = (float*)d_out;           // [G,P,1,64,64] fp32

    dim3 grid(G_ * P_);   // 1536 pairs, one workgroup each
    dim3 block(256);      // 8 waves (wave32)
    hipLaunchKernelGGL(gcc_fused_wmma_kernel, grid, block, 0, stream,
                       x, sizes, out);
}